// F1Score_19164144075466
// MI455X (gfx1250) — compile-verified
//
#include <hip/hip_runtime.h>

typedef float v2f __attribute__((ext_vector_type(2)));
typedef float v8f __attribute__((ext_vector_type(8)));

#define N_PTS   8192
#define CHUNK   2048          // points of the "column" array staged in LDS (32KB as float4)
#define THRESH  0.0001f

// One directional chamfer pass. Each wave owns 16 rows of P, loops over all
// 8192 points of Q in LDS-staged chunks, computes 16x16 distance tiles with
// V_WMMA_F32_16X16X4_F32, keeps a register-resident running min, and finally
// counts rows whose min distance is below THRESH.
//
// grid.x = 4 (batch*dir) * 64 blocks; block = 256 threads = 8 waves.
__global__ __launch_bounds__(256) void chamfer_dir_kernel(
    const float* __restrict__ array1,
    const float* __restrict__ array2,
    unsigned int* __restrict__ counts)   // counts[b*2 + dir]
{
    __shared__ float4 sB[CHUNK];         // (y0, y1, y2, ||y||^2)

    const int tid   = threadIdx.x;
    const int lane  = tid & 31;
    const int wave  = tid >> 5;
    const int col16 = lane & 15;
    const bool hi   = (lane >= 16);

    const int bd  = blockIdx.x >> 6;     // 0..3 : {batch, dir}
    const int blk = blockIdx.x & 63;     // 0..63
    const int b   = bd >> 1;
    const int dir = bd & 1;

    const float* P = (dir == 0) ? array1 : array2;   // rows  (min over Q)
    const float* Q = (dir == 0) ? array2 : array1;   // cols
    const float* pb = P + (size_t)b * N_PTS * 3;
    const float* qb = Q + (size_t)b * N_PTS * 3;

    // ---- A strip: 16 rows per wave, augmented a = (-2x0,-2x1,-2x2, 1) ----
    // 32-bit A 16x4 layout: lanes 0-15 hold K=0,1 (VGPR0,1); lanes 16-31 hold K=2,3.
    const int strip = blk * 8 + wave;                // 0..511
    const int row   = strip * 16 + col16;
    const float ax = pb[row * 3 + 0];
    const float ay = pb[row * 3 + 1];
    const float az = pb[row * 3 + 2];
    const float sq1 = ax * ax + ay * ay + az * az;

    v2f A;
    A.x = hi ? (-2.0f * az) : (-2.0f * ax);          // K=0 / K=2
    A.y = hi ? 1.0f         : (-2.0f * ay);          // K=1 / K=3

    // ---- C init: C[v][lane] = ||x_row||^2, row = v + (hi ? 8 : 0). Loop invariant. ----
    v8f C;
#pragma unroll
    for (int v = 0; v < 8; ++v) {
        C[v] = __shfl(sq1, v + (hi ? 8 : 0), 32);    // ds_bpermute_b32
    }

    v8f minD;
#pragma unroll
    for (int v = 0; v < 8; ++v) minD[v] = 3.0e38f;

    for (int chunk = 0; chunk < N_PTS / CHUNK; ++chunk) {
        __syncthreads();
        // Cooperative stage of Q chunk into LDS with precomputed ||y||^2.
        for (int i = tid; i < CHUNK; i += 256) {
            const float* g = qb + (size_t)(chunk * CHUNK + i) * 3;
            const float x = g[0], y = g[1], z = g[2];
            sB[i] = make_float4(x, y, z, x * x + y * y + z * z);
        }
        __syncthreads();

        for (int t = 0; t < CHUNK / 16; ++t) {
            const float4 p = sB[t * 16 + col16];     // ds_load_b128, broadcast across halves
            // B 4x16 layout: lanes 0-15 hold K=0,1; lanes 16-31 hold K=2,3.
            // b = (y0, y1, y2, ||y||^2)
            v2f B;
            B.x = hi ? p.z : p.x;                    // K=0 / K=2
            B.y = hi ? p.w : p.y;                    // K=1 / K=3

            // D[r][c] = ||x_r||^2 - 2 x_r . y_c + ||y_c||^2
            v8f D = __builtin_amdgcn_wmma_f32_16x16x4_f32(
                false, A, false, B, (short)0, C, false, false);

#pragma unroll
            for (int v = 0; v < 8; ++v)
                minD[v] = fminf(minD[v], D[v]);
        }
    }

    // ---- Row-min cross-lane reduction (within each 16-lane half) + count ----
    unsigned int cnt = 0;
#pragma unroll
    for (int v = 0; v < 8; ++v) {
        float r = minD[v];
        r = fminf(r, __shfl_xor(r, 1, 32));
        r = fminf(r, __shfl_xor(r, 2, 32));
        r = fminf(r, __shfl_xor(r, 4, 32));
        r = fminf(r, __shfl_xor(r, 8, 32));
        // lanes 0-15 now hold rowmin of row v; lanes 16-31 rowmin of row v+8.
        unsigned long long bal = __ballot(r < THRESH);
        cnt += __popcll(bal & 0x10001ull);           // one representative bit per half
    }
    if (lane == 0) atomicAdd(&counts[b * 2 + dir], cnt);
}

__global__ void f1_init_kernel(unsigned int* counts) {
    if (threadIdx.x < 4) counts[threadIdx.x] = 0u;
}

__global__ void f1_final_kernel(const unsigned int* __restrict__ counts,
                                float* __restrict__ out) {
    if (threadIdx.x == 0) {
#pragma unroll
        for (int b = 0; b < 2; ++b) {
            const float p1 = (float)counts[b * 2 + 0] * (1.0f / (float)N_PTS);
            const float p2 = (float)counts[b * 2 + 1] * (1.0f / (float)N_PTS);
            const float den = p1 + p2;
            const float f = (den > 0.0f) ? (2.0f * p1 * p2 / den) : 0.0f;
            out[0 + b] = f;    // fscore[2]
            out[2 + b] = p1;   // precision_1[2]
            out[4 + b] = p2;   // precision_2[2]
        }
    }
}

extern "C" void kernel_launch(void* const* d_in, const int* in_sizes, int n_in,
                              void* d_out, int out_size, void* d_ws, size_t ws_size,
                              hipStream_t stream) {
    const float* array1 = (const float*)d_in[0];   // [2, 8192, 3] f32
    const float* array2 = (const float*)d_in[1];   // [2, 8192, 3] f32
    unsigned int* counts = (unsigned int*)d_ws;    // 4 uints: [b][dir]
    float* out = (float*)d_out;                    // 6 floats

    f1_init_kernel<<<1, 32, 0, stream>>>(counts);
    // 2 batches * 2 directions * 64 blocks (8 waves x 16-row strips each)
    chamfer_dir_kernel<<<256, 256, 0, stream>>>(array1, array2, counts);
    f1_final_kernel<<<1, 32, 0, stream>>>(counts, out);
}